// MechanisticForward_12618613915678
// MI455X (gfx1250) — compile-verified
//
#include <hip/hip_runtime.h>
#include <hip/hip_bf16.h>
#include <math.h>

#define KDIM 4096
#define BDIM 50257
#define CDIM 2048
#define WGEOM 0.55f
#define WLM   0.30f
#define WBAR  0.15f
#define EPSF  1e-12f
#define WAVES 4        // waves per block; each wave owns 16 candidates
#define KSEG  32       // split-K segments -> 4096 independent wave-jobs
#define KCH   (KDIM / KSEG)   // 128 rows per segment

typedef float v2f __attribute__((ext_vector_type(2)));
typedef float v8f __attribute__((ext_vector_type(8)));

// ---------------------------------------------------------------------------
// Phase A: proto[k] = (sum over valid ctx of W[k, ctx]) / max(used,1)
// ---------------------------------------------------------------------------
__global__ void proto_kernel(const int* __restrict__ ctx,
                             const float* __restrict__ W,
                             float* __restrict__ proto,
                             int* __restrict__ used_out) {
  int k = blockIdx.x * blockDim.x + threadIdx.x;
  if (k >= KDIM) return;
  float s = 0.f;
  int used = 0;
#pragma unroll
  for (int t = 0; t < 3; ++t) {
    int c = ctx[t];
    bool m = (c >= 0) && (c < BDIM);
    int cc = m ? c : 0;
    float w = W[(long long)k * BDIM + cc];
    s += m ? w : 0.f;
    used += m ? 1 : 0;
  }
  proto[k] = s / (float)(used > 0 ? used : 1);
  if (k == 0) *used_out = used;
}

// ---------------------------------------------------------------------------
// Phase B: npr = sqrt(sum proto^2) + EPS   (single block)
// ---------------------------------------------------------------------------
__global__ void npr_kernel(const float* __restrict__ proto,
                           float* __restrict__ npr_out) {
  __shared__ float red[8];
  float s = 0.f;
  for (int i = threadIdx.x; i < KDIM; i += blockDim.x) {
    float p = proto[i];
    s += p * p;
  }
  for (int off = 16; off; off >>= 1) s += __shfl_down(s, off, 32);
  int lane = threadIdx.x & 31, wv = threadIdx.x >> 5;
  if (lane == 0) red[wv] = s;
  __syncthreads();
  if (threadIdx.x == 0) {
    float t = 0.f;
    for (int i = 0; i < 8; ++i) t += red[i];
    *npr_out = sqrtf(t) + EPSF;
  }
}

// ---------------------------------------------------------------------------
// Phase C (split-K partials): each wave owns (16 candidates) x (128 K-rows).
// V_WMMA_F32_16X16X4_F32 per 4-row chunk:
//   c_dots += A(proto row0) x B(V chunk)  -> row 0 = partial dots
//   c_gram += A(V chunk)    x B(V chunk)  -> diagonal = partial ||V_j||^2
// One gathered float2 per lane feeds both WMMAs (A/B per-lane images match).
// Partials go to accDN[seg][j] = {dot, nv2}; no atomics -> deterministic.
// Block mapping: blockIdx.x = tileGroup * KSEG + seg, so all 4 waves of a
// block share one K-segment and its 512-byte proto stage.
// ---------------------------------------------------------------------------
__global__ void partial_kernel(const float* __restrict__ W,
                               const float* __restrict__ proto_g,
                               const int* __restrict__ cand_idx,
                               float* __restrict__ accDN) {
  __shared__ float sproto[KCH];             // 512 B
  __shared__ float sgram[WAVES][32][8];     // 4 KB
  const int seg = blockIdx.x & (KSEG - 1);
  const int tg  = blockIdx.x / KSEG;        // tile group (4 tiles)
  const int kg0 = seg * KCH;

  if ((int)threadIdx.x < KCH) sproto[threadIdx.x] = proto_g[kg0 + threadIdx.x];
  __syncthreads();

  const int lane = threadIdx.x & 31;
  const int wv   = threadIdx.x >> 5;
  const int jn   = (tg * WAVES + wv) * 16 + (lane & 15);
  const int ci   = cand_idx[jn];
  const int colc = ci < 0 ? 0 : (ci >= BDIM ? BDIM - 1 : ci);  // jnp.clip
  const int h    = lane >> 4;               // lane half selects K pair
  const bool isrow0 = (lane & 15) == 0;     // A row 0 carries proto

  v8f cd = {};   // dots accumulator
  v8f cg = {};   // gram accumulator (diagonal = norms^2)

  // main loop: prefetch 16 rows ahead (both row lines), strictly in-segment,
  // so addresses are provably in-bounds -> safe at near scope
  int kl = 0;
#pragma unroll 4
  for (; kl < KCH - 16; kl += 4) {
    const long long base = (long long)(kg0 + kl + 2 * h) * BDIM + colc;
    v2f v;
    v.x = W[base];
    v.y = W[base + BDIM];
    __builtin_prefetch(&W[base + 16LL * BDIM], 0, 3);  // future v.x line
    __builtin_prefetch(&W[base + 17LL * BDIM], 0, 3);  // future v.y line
    v2f p = *(const v2f*)&sproto[kl + 2 * h];
    v2f a;
    a.x = isrow0 ? p.x : 0.f;
    a.y = isrow0 ? p.y : 0.f;
    cd = __builtin_amdgcn_wmma_f32_16x16x4_f32(false, a, false, v, (short)0,
                                               cd, false, false);
    cg = __builtin_amdgcn_wmma_f32_16x16x4_f32(false, v, false, v, (short)0,
                                               cg, false, false);
  }
  // tail: last 4 chunks, no prefetch, branch-free
#pragma unroll
  for (; kl < KCH; kl += 4) {
    const long long base = (long long)(kg0 + kl + 2 * h) * BDIM + colc;
    v2f v;
    v.x = W[base];
    v.y = W[base + BDIM];
    v2f p = *(const v2f*)&sproto[kl + 2 * h];
    v2f a;
    a.x = isrow0 ? p.x : 0.f;
    a.y = isrow0 ? p.y : 0.f;
    cd = __builtin_amdgcn_wmma_f32_16x16x4_f32(false, a, false, v, (short)0,
                                               cd, false, false);
    cg = __builtin_amdgcn_wmma_f32_16x16x4_f32(false, v, false, v, (short)0,
                                               cg, false, false);
  }

  // bounce gram D through LDS to pull diagonal entries to the owning lane
#pragma unroll
  for (int r = 0; r < 8; ++r) sgram[wv][lane][r] = cg[r];
  __syncthreads();

  if (lane < 16) {
    // D[n][n]: n<8 -> VGPR n, lane n ; n>=8 -> VGPR n-8, lane n+16
    float nv2 = (lane < 8) ? sgram[wv][lane][lane]
                           : sgram[wv][lane + 16][lane - 8];
    v2f dn;
    dn.x = cd[0];   // D[0][lane] = partial dot
    dn.y = nv2;     // partial squared norm
    *(v2f*)&accDN[2LL * ((long long)seg * CDIM + jn)] = dn;  // b64 store
  }
}

// ---------------------------------------------------------------------------
// Phase D: fixed-order segment reduction + scoring (deterministic)
// ---------------------------------------------------------------------------
__global__ void finalize_kernel(const float* __restrict__ accDN,
                                const float* __restrict__ npr_p,
                                const int* __restrict__ used_p,
                                const float* __restrict__ cand_probs,
                                const int* __restrict__ cand_idx,
                                const float* __restrict__ bar_probs,
                                float* __restrict__ scores) {
  int j = blockIdx.x * blockDim.x + threadIdx.x;
  if (j >= CDIM) return;
  float d = 0.f, n2 = 0.f;
#pragma unroll 8
  for (int s = 0; s < KSEG; ++s) {
    v2f dn = *(const v2f*)&accDN[2LL * ((long long)s * CDIM + j)];
    d += dn.x;
    n2 += dn.y;
  }
  int ci = cand_idx[j];
  bool valid = (ci >= 0) && (ci < BDIM);
  int colc = ci < 0 ? 0 : (ci >= BDIM ? BDIM - 1 : ci);
  float nv   = sqrtf(n2) + EPSF;
  float npr  = *npr_p;
  int   used = *used_p;
  float geom = (used > 0) ? (d / (npr * nv)) : 0.f;
  float cp   = cand_probs[j];
  scores[j] = valid ? (WGEOM * geom + WLM * cp + WBAR * bar_probs[colc]) : cp;
}

// ---------------------------------------------------------------------------
// Phase E: argmax over 2048 scores, first-index tie-break (jnp.argmax)
// ---------------------------------------------------------------------------
__global__ void argmax_kernel(const float* __restrict__ scores,
                              int* __restrict__ out_idx) {
  __shared__ float bv[256];
  __shared__ int   bi[256];
  float best = -3.402823466e38f;
  int besti = 0;
  for (int i = threadIdx.x; i < CDIM; i += blockDim.x) {
    float s = scores[i];
    if (s > best) { best = s; besti = i; }
  }
  bv[threadIdx.x] = best;
  bi[threadIdx.x] = besti;
  __syncthreads();
  for (int s2 = 128; s2; s2 >>= 1) {
    if ((int)threadIdx.x < s2) {
      float ov = bv[threadIdx.x + s2];
      int   oi = bi[threadIdx.x + s2];
      if (ov > bv[threadIdx.x] ||
          (ov == bv[threadIdx.x] && oi < bi[threadIdx.x])) {
        bv[threadIdx.x] = ov;
        bi[threadIdx.x] = oi;
      }
    }
    __syncthreads();
  }
  if (threadIdx.x == 0) *out_idx = bi[0];
}

// ---------------------------------------------------------------------------
extern "C" void kernel_launch(void* const* d_in, const int* in_sizes, int n_in,
                              void* d_out, int out_size, void* d_ws,
                              size_t ws_size, hipStream_t stream) {
  const int*   ctx  = (const int*)d_in[0];    // context_idx (3)
  const float* W    = (const float*)d_in[1];  // binding_W (K*B)
  const float* bar  = (const float*)d_in[2];  // bar_probs (B)
  const float* cp   = (const float*)d_in[3];  // cand_probs (C)
  const int*   cidx = (const int*)d_in[4];    // cand_idx (C)

  float* proto = (float*)d_ws;                // KDIM floats
  float* npr   = proto + KDIM;                // 1 float
  int*   used  = (int*)(proto + KDIM + 1);    // 1 int
  float* accDN = proto + KDIM + 2;            // KSEG*CDIM*2 floats (512 KB)

  int*   argout = (int*)d_out;                // output[0]: argmax (int32 bits)
  float* scores = (float*)d_out + 1;          // output[1..C]: scores

  proto_kernel<<<KDIM / 256, 256, 0, stream>>>(ctx, W, proto, used);
  npr_kernel<<<1, 256, 0, stream>>>(proto, npr);
  // (CDIM/16/WAVES) tile groups x KSEG segments = 32 * 32 = 1024 blocks
  partial_kernel<<<(CDIM / (16 * WAVES)) * KSEG, 32 * WAVES, 0, stream>>>(
      W, proto, cidx, accDN);
  finalize_kernel<<<CDIM / 256, 256, 0, stream>>>(accDN, npr, used, cp, cidx,
                                                  bar, scores);
  argmax_kernel<<<1, 256, 0, stream>>>(scores, argout);
}